// GraphAttentionHead_68745246540453
// MI455X (gfx1250) — compile-verified
//
#include <hip/hip_runtime.h>
#include <hip/hip_bf16.h>

typedef __attribute__((ext_vector_type(2))) float v2f;
typedef __attribute__((ext_vector_type(8))) float v8f;

#define NODE_DIM 64
#define EDGE_DIM 16
#define GRAPH_IN 32
#define GOUT 64
#define ALPHA 0.2f

// ---------------------------------------------------------------------------
// Prep: fold parameters.
//   w_node[in]  = sum_o W_node[in*64+o] * a_node[64+o]      (64 floats)
//   w_edge[in]  = sum_o W_edge[in*64+o] * a_edge[64+o]      (16 floats)
//   b_node      = sum_o (sum_k graph[k]*W_graph[k*64+o]) * a_node[o]
//   b_edge      = same with a_edge
// ws float layout: [0,64) w_node | [64,80) w_edge | [80] b_node | [81] b_edge
// ---------------------------------------------------------------------------
__global__ void gat_prep_kernel(const float* __restrict__ graph_fts,
                                const float* __restrict__ W_node,
                                const float* __restrict__ W_edge,
                                const float* __restrict__ W_graph,
                                const float* __restrict__ a_node,
                                const float* __restrict__ a_edge,
                                float* __restrict__ wsf) {
    int t = threadIdx.x;
    if (t < NODE_DIM) {
        float acc = 0.f;
        #pragma unroll 8
        for (int o = 0; o < GOUT; ++o)
            acc = fmaf(W_node[t * GOUT + o], a_node[GOUT + o], acc);
        wsf[t] = acc;
    }
    if (t < EDGE_DIM) {
        float acc = 0.f;
        #pragma unroll 8
        for (int o = 0; o < GOUT; ++o)
            acc = fmaf(W_edge[t * GOUT + o], a_edge[GOUT + o], acc);
        wsf[NODE_DIM + t] = acc;
    }
    if (t == 0) {
        float bn = 0.f, be = 0.f;
        for (int o = 0; o < GOUT; ++o) {
            float g = 0.f;
            #pragma unroll 8
            for (int k = 0; k < GRAPH_IN; ++k)
                g = fmaf(graph_fts[k], W_graph[k * GOUT + o], g);
            bn = fmaf(g, a_node[o], bn);
            be = fmaf(g, a_edge[o], be);
        }
        wsf[NODE_DIM + EDGE_DIM]     = bn;
        wsf[NODE_DIM + EDGE_DIM + 1] = be;
    }
}

// ---------------------------------------------------------------------------
// WMMA GEMV: one wave processes 16 rows of fts[R,K] (K multiple of 4).
// A operand: weight vector w broadcast to all 16 M-rows.
//   A 16x4 f32 layout: lane<16 -> (w[k], w[k+1]); lane>=16 -> (w[k+2], w[k+3])
// B operand: 16 feature rows as columns.
//   B 4x16 f32 layout: lane L holds col N=L&15; VGPRs give K = k+2*(L>>4)..+1
// D[m][n] = dot(w, row n) for every m -> C VGPR0, lanes 0..15 hold the results.
// Epilogue (lanes 0..15): att = exp(clip(leaky_relu(bias + dot))),
//   store to att_out, atomicAdd into per-segment sum.
// ---------------------------------------------------------------------------
__global__ void gat_att_wmma_kernel(const float* __restrict__ fts,
                                    const int* __restrict__ seg,
                                    const float* __restrict__ wsf,
                                    int w_off, int b_off,
                                    float* __restrict__ att_out,
                                    float* __restrict__ sums,
                                    int R, int K) {
    const int lane = threadIdx.x & 31;
    const long wave = (long)((blockIdx.x * blockDim.x + threadIdx.x) >> 5);
    const long rowBase = wave * 16;
    if (rowBase >= R) return;              // wave-uniform: EXEC stays all-ones

    const int hi  = lane >> 4;             // 0 or 1
    const int col = lane & 15;
    const float* __restrict__ w = wsf + w_off;
    long r = rowBase + col;
    long rl = r < (long)R ? r : (long)R - 1;   // clamp load row (R%16==0 anyway)
    const float* __restrict__ rowp = fts + rl * (long)K;

    v8f c = {0.f, 0.f, 0.f, 0.f, 0.f, 0.f, 0.f, 0.f};
    for (int k = 0; k < K; k += 4) {
        v2f a, b;
        a.x = w[k + 2 * hi];
        a.y = w[k + 2 * hi + 1];
        const float* p = rowp + k + 2 * hi;
        b.x = p[0];
        b.y = p[1];
        // D = A x B + C : emits v_wmma_f32_16x16x4_f32
        c = __builtin_amdgcn_wmma_f32_16x16x4_f32(
                /*neg_a=*/false, a, /*neg_b=*/false, b,
                /*c_mod=*/(short)0, c, /*reuse_a=*/false, /*reuse_b=*/false);
    }

    if (hi == 0 && r < (long)R) {
        float x = wsf[b_off] + c[0];
        x = x > 0.f ? x : ALPHA * x;                 // leaky_relu
        x = fminf(2.f, fmaxf(-2.f, x));              // clip
        float e = __expf(x);
        att_out[r] = e;
        atomicAdd(&sums[seg[r]], e);
    }
}

// ---------------------------------------------------------------------------
// Normalize both halves of d_out by gathered segment sums.
// ---------------------------------------------------------------------------
__global__ void gat_norm_kernel(const int* __restrict__ seg,
                                const float* __restrict__ sums_n,
                                const float* __restrict__ sums_e,
                                float* __restrict__ out, int N) {
    int i = blockIdx.x * blockDim.x + threadIdx.x;
    if (i < N) {
        int s = seg[i];
        out[i]     = out[i]     / sums_n[s];
        out[N + i] = out[N + i] / sums_e[s];
    }
}

extern "C" void kernel_launch(void* const* d_in, const int* in_sizes, int n_in,
                              void* d_out, int out_size, void* d_ws, size_t ws_size,
                              hipStream_t stream) {
    const float* node_fts  = (const float*)d_in[0];   // [N,64]
    // d_in[1] = edge_fts (unused by the reference computation)
    const float* graph_fts = (const float*)d_in[2];   // [1,32]
    const float* edges     = (const float*)d_in[3];   // [E,16]  (used for e_v!)
    const int*   seg_ids   = (const int*)  d_in[4];   // [E]
    const float* W_node    = (const float*)d_in[5];   // [64,64]
    const float* W_edge    = (const float*)d_in[6];   // [16,64]
    const float* W_graph   = (const float*)d_in[7];   // [32,64]
    const float* a_node    = (const float*)d_in[8];   // [128,1]
    const float* a_edge    = (const float*)d_in[9];   // [128,1]

    const int N = in_sizes[0] / NODE_DIM;             // 1,000,000
    const int E = in_sizes[4];                        // 1,000,000
    float* out = (float*)d_out;                       // [N + E]

    // ws layout (floats): [0,64) w_node | [64,80) w_edge | [80] b_n | [81] b_e
    //                     | pad to 128 | [128,128+N) sums_n | [...,+E) sums_e
    float* wsf    = (float*)d_ws;
    float* sums_n = wsf + 128;
    float* sums_e = sums_n + N;

    // 1) zero segment sums (capture-safe async memset)
    hipMemsetAsync((void*)sums_n, 0, (size_t)(N + E) * sizeof(float), stream);

    // 2) fold parameters
    gat_prep_kernel<<<1, 64, 0, stream>>>(graph_fts, W_node, W_edge, W_graph,
                                          a_node, a_edge, wsf);

    // 3) node attention (K=64): 16 rows per wave
    {
        long waves = (N + 15) / 16;
        long threads = waves * 32;
        int blocks = (int)((threads + 255) / 256);
        gat_att_wmma_kernel<<<blocks, 256, 0, stream>>>(
            node_fts, seg_ids, wsf, /*w_off=*/0, /*b_off=*/80,
            out, sums_n, N, NODE_DIM);
    }

    // 4) edge attention (K=16) — uses `edges`, matching the reference
    {
        long waves = (E + 15) / 16;
        long threads = waves * 32;
        int blocks = (int)((threads + 255) / 256);
        gat_att_wmma_kernel<<<blocks, 256, 0, stream>>>(
            edges, seg_ids, wsf, /*w_off=*/64, /*b_off=*/81,
            out + N, sums_e, E, EDGE_DIM);
    }

    // 5) segment-softmax normalization for both outputs
    {
        int blocks = (N + 255) / 256;
        gat_norm_kernel<<<blocks, 256, 0, stream>>>(seg_ids, sums_n, sums_e, out, N);
    }
}